// CityExpertRouter_81561428951525
// MI455X (gfx1250) — compile-verified
//
#include <hip/hip_runtime.h>
#include <hip/hip_bf16.h>
#include <math.h>

#define EMBED      2048
#define NEXP       16
#define TOKENS_PER_WAVE 16
#define LDS_STRIDE 2052   // 2048 + 4 dword pad -> conflict-free ds_load_b64 across 16 lanes

typedef __attribute__((ext_vector_type(2))) float v2f;
typedef __attribute__((ext_vector_type(4))) float v4f;
typedef __attribute__((ext_vector_type(8))) float v8f;

// One wave handles a 16-token x 16-expert tile:
//   D[16x16] += A[16x4] * B[4x16] over K=2048 via v_wmma_f32_16x16x4_f32.
// gate_w is staged in LDS once per block (it thrashes the 64KB WGP$ otherwise).
__global__ __launch_bounds__(256) void
CityExpertRouter_kernel(const float* __restrict__ x,
                        const float* __restrict__ gate_w,
                        float* __restrict__ out_w,
                        float* __restrict__ out_i,
                        int n_tiles)
{
    extern __shared__ float sgw[];   // [NEXP][LDS_STRIDE]

    // ---- Stage gate_w (16 x 2048 f32 = 128KB) into padded LDS, b128 at a time ----
    for (int t = threadIdx.x; t < NEXP * (EMBED / 4); t += blockDim.x) {
        const int e  = t / (EMBED / 4);
        const int k4 = t % (EMBED / 4);
        v4f v = *(const v4f*)(gate_w + (size_t)e * EMBED + k4 * 4);
        *(v4f*)(&sgw[e * LDS_STRIDE + k4 * 4]) = v;
    }
    __syncthreads();

    const int lane = threadIdx.x & 31;
    const int tile = blockIdx.x * (blockDim.x >> 5) + (threadIdx.x >> 5);
    if (tile >= n_tiles) return;          // wave-uniform branch

    const int  half = lane >> 4;          // K sub-offset select (ISA A/B layout)
    const int  r    = lane & 15;          // token row (A) / expert col (B)
    const long tokenBase = (long)tile * TOKENS_PER_WAVE;

    const float* ap = x + (tokenBase + r) * (long)EMBED + half * 2;   // A: x rows
    const float* bp = &sgw[r * LDS_STRIDE + half * 2];                // B: gate_w rows (LDS)

    v8f c = {0.f, 0.f, 0.f, 0.f, 0.f, 0.f, 0.f, 0.f};
#pragma unroll 8
    for (int k = 0; k < EMBED; k += 4) {
        v2f a = *(const v2f*)(ap + k);    // global_load_b64, streams one token row/lane
        v2f b = *(const v2f*)(bp + k);    // ds_load_b64, conflict-free
        c = __builtin_amdgcn_wmma_f32_16x16x4_f32(
                /*neg_a=*/false, a, /*neg_b=*/false, b,
                /*c_mod=*/(short)0, c, /*reuse_a=*/false, /*reuse_b=*/false);
    }

    // C/D layout: VGPR j, lanes 0-15 -> token M=j (N=lane); lanes 16-31 -> token M=j+8.
    // Each token's 16 expert logits sit one-per-lane in a 16-lane half of one register.
    // Top-2 with tie -> lower index (matches jax.lax.top_k), then
    // renormalized softmax weights reduce to: w1 = 1/(1+exp(l2-l1)), w2 = 1-w1.
#pragma unroll
    for (int j = 0; j < 8; ++j) {
        float t1 = c[j]; int i1 = r;
        float t2 = -3.0e38f; int i2 = NEXP;
#pragma unroll
        for (int m = 8; m >= 1; m >>= 1) {
            float o1  = __shfl_xor(t1, m, 32);
            int   oi1 = __shfl_xor(i1, m, 32);
            float o2  = __shfl_xor(t2, m, 32);
            int   oi2 = __shfl_xor(i2, m, 32);
            // merge two sorted top-2 lists
            bool  sw  = (o1 > t1) || (o1 == t1 && oi1 < i1);
            float w1v = sw ? o1 : t1;  int w1i = sw ? oi1 : i1;
            float lv  = sw ? t1 : o1;  int li  = sw ? i1  : oi1;   // loser of firsts
            bool  s2  = (o2 > t2) || (o2 == t2 && oi2 < i2);
            float bsv = s2 ? o2 : t2;  int bsi = s2 ? oi2 : i2;    // best of seconds
            bool  s3  = (bsv > lv) || (bsv == lv && bsi < li);
            t1 = w1v; i1 = w1i;
            t2 = s3 ? bsv : lv;  i2 = s3 ? bsi : li;
        }
        if (r == 0) {
            const long token = tokenBase + j + 8 * half;
            const float w1 = 1.0f / (1.0f + expf(t2 - t1));
            out_w[token * 2 + 0] = w1;
            out_w[token * 2 + 1] = 1.0f - w1;
            out_i[token * 2 + 0] = (float)i1;
            out_i[token * 2 + 1] = (float)i2;
        }
    }
}

extern "C" void kernel_launch(void* const* d_in, const int* in_sizes, int n_in,
                              void* d_out, int out_size, void* d_ws, size_t ws_size,
                              hipStream_t stream) {
    const float* x  = (const float*)d_in[0];   // [8,4096,2048] f32
    const float* gw = (const float*)d_in[1];   // [16,2048]     f32

    const int n_tokens = in_sizes[0] / EMBED;              // 32768
    const int n_tiles  = n_tokens / TOKENS_PER_WAVE;       // 2048 waves

    float* out_w = (float*)d_out;                          // [n_tokens,2] weights
    float* out_i = out_w + (size_t)n_tokens * 2;           // [n_tokens,2] indices (as float)

    const int waves_per_block = 8;                         // 256 threads
    const int blocks = (n_tiles + waves_per_block - 1) / waves_per_block;
    const size_t lds_bytes = (size_t)NEXP * LDS_STRIDE * sizeof(float);  // ~132KB

    CityExpertRouter_kernel<<<blocks, waves_per_block * 32, lds_bytes, stream>>>(
        x, gw, out_w, out_i, n_tiles);
}